// AttCNN4All_7911329759993
// MI455X (gfx1250) — compile-verified
//
#include <hip/hip_runtime.h>
#include <float.h>

// Problem constants (from reference)
#define KV_LEN 2048
#define BATCH  64
#define QS     256          // Q_SIZE = K_SIZE = V_SIZE
#define KW     3
#define RDIM   (KW * QS)    // 768; reduction index r = w*256 + c

typedef __attribute__((ext_vector_type(2))) float v2f;
typedef __attribute__((ext_vector_type(8))) float v8f;

// CDNA5 fp32 matrix op: D(16x16) = A(16x4) * B(4x16) + C
// 8-arg form: (neg_a, A, neg_b, B, c_mod, C, reuse_a, reuse_b)
__device__ __forceinline__ v8f wmma_f32_16x16x4(v2f a, v2f b, v8f c) {
  return __builtin_amdgcn_wmma_f32_16x16x4_f32(false, a, false, b, (short)0, c,
                                               false, false);
}

// -----------------------------------------------------------------------------
// Kernel 1: Kern[b][o][r] = sum_kq q[b,kq] * W[n,kq] + bias[n]
//   n = 768*o + 3*c + w ; r = w*256 + c  (so GEMM2 reads Kern contiguously in r)
// Grid (3 w, 256 o); block = 256 threads = 8 wave32.
// Per block: M = 64 (batch) x N = 256 (c) tile, K = 256.
// Wave wv: M-tile = wv&3 (16 rows), N-half = wv>>2 (128 cols -> 8 fragments).
// -----------------------------------------------------------------------------
__global__ __launch_bounds__(256) void k1_qw(const float* __restrict__ q,
                                             const float* __restrict__ W,
                                             const float* __restrict__ bias,
                                             float* __restrict__ kern) {
  const int w    = blockIdx.x;        // 0..2
  const int o    = blockIdx.y;        // 0..255
  const int lane = threadIdx.x & 31;
  const int wv   = threadIdx.x >> 5;
  const int mt   = wv & 3;            // which 16-row batch tile
  const int ng   = wv >> 2;           // which 128-col half
  const int l16  = lane & 15;
  const int lhi  = lane >> 4;         // 0/1 -> K offset 0 / 2

  const int  m0 = mt * 16;
  const int  c0 = ng * 128;
  const long n0 = 768L * o + w;       // W row for c = 0

  v8f acc[8];
#pragma unroll
  for (int f = 0; f < 8; ++f)
    acc[f] = (v8f){0.f, 0.f, 0.f, 0.f, 0.f, 0.f, 0.f, 0.f};

  for (int kq = 0; kq < QS; kq += 4) {
    const int kb = kq + 2 * lhi;
    // A fragment (16x4 of q): lane holds q[m0+l16, kb..kb+1]
    const v2f a = *reinterpret_cast<const v2f*>(q + (long)(m0 + l16) * QS + kb);
#pragma unroll
    for (int f = 0; f < 8; ++f) {
      // B fragment (4x16 of W^T): column cidx -> W row n0 + 3*cidx
      const long n = n0 + 3L * (c0 + f * 16 + l16);
      const v2f bf = *reinterpret_cast<const v2f*>(W + n * QS + kb);
      acc[f] = wmma_f32_16x16x4(a, bf, acc[f]);
    }
  }

  // Epilogue: add bias, coalesced store into Kern[b][o][r]
#pragma unroll
  for (int f = 0; f < 8; ++f) {
    const int   c  = c0 + f * 16 + l16;
    const float bv = bias[n0 + 3L * c];
    const long  r  = (long)w * 256 + c;
#pragma unroll
    for (int j = 0; j < 8; ++j) {
      const int bb = m0 + j + 8 * lhi;  // batch row (C layout: M=j / j+8)
      kern[((long)bb * QS + o) * RDIM + r] = acc[f][j] + bv;
    }
  }
}

// -----------------------------------------------------------------------------
// Kernel 2: a_ij[t,b,o] = sum_r Win[t,r] * Kern[b][o][r],
//   Win[t, w*256+c] = k[t+w-1, b, c]  (zero-padded in t)
// Grid (16 t-tiles, 64 b); block = 256 threads = 8 wave32.
// Tile: M = 128 (t) x N = 256 (o), K = 768.
// Wave wv: t-sub = (wv&3)*32 (2 M-frags), o-half = (wv>>2)*128 (8 N-frags).
// 16 f32 accumulators (128 VGPRs/lane).
// -----------------------------------------------------------------------------
__global__ __launch_bounds__(256) void k2_attn(const float* __restrict__ kmat,
                                               const float* __restrict__ kern,
                                               float* __restrict__ aij) {
  const int tt   = blockIdx.x;
  const int b    = blockIdx.y;
  const int lane = threadIdx.x & 31;
  const int wv   = threadIdx.x >> 5;
  const int mg   = wv & 3;
  const int ng   = wv >> 2;
  const int l16  = lane & 15;
  const int lhi  = lane >> 4;

  const int t0 = tt * 128 + mg * 32;
  const int o0 = ng * 128;

  v8f acc[2][8];
#pragma unroll
  for (int mt = 0; mt < 2; ++mt)
#pragma unroll
    for (int nf = 0; nf < 8; ++nf)
      acc[mt][nf] = (v8f){0.f, 0.f, 0.f, 0.f, 0.f, 0.f, 0.f, 0.f};

  const float* kernb = kern + ((long)b * QS + o0) * RDIM;

  for (int w = 0; w < KW; ++w) {
    for (int kk = 0; kk < 256; kk += 4) {
      const int c2 = kk + 2 * lhi;            // c base for this lane (same w seg)
      // A fragments: Win[t0+mt*16+l16, r..r+1] = k[t+w-1, b, c2..c2+1]
      v2f afr[2];
#pragma unroll
      for (int mt = 0; mt < 2; ++mt) {
        const int ts = t0 + mt * 16 + l16 + w - 1;
        v2f av = (v2f){0.f, 0.f};
        if ((unsigned)ts < (unsigned)KV_LEN)
          av = *reinterpret_cast<const v2f*>(kmat +
                   ((long)ts * BATCH + b) * QS + c2);
        afr[mt] = av;
      }
      const int rb = w * 256 + c2;
#pragma unroll
      for (int nf = 0; nf < 8; ++nf) {
        // B fragment: Kern[b][o0+nf*16+l16][rb..rb+1] (contiguous in r)
        const v2f bf = *reinterpret_cast<const v2f*>(
            kernb + (long)(nf * 16 + l16) * RDIM + rb);
#pragma unroll
        for (int mt = 0; mt < 2; ++mt)
          acc[mt][nf] = wmma_f32_16x16x4(afr[mt], bf, acc[mt][nf]);
      }
    }
  }

  // Store a_ij[t,b,o]; lanes 0..15 -> consecutive o -> 64B coalesced stores
#pragma unroll
  for (int mt = 0; mt < 2; ++mt)
#pragma unroll
    for (int j = 0; j < 8; ++j) {
      const int t = t0 + mt * 16 + j + 8 * lhi;
#pragma unroll
      for (int nf = 0; nf < 8; ++nf) {
        const int o = o0 + nf * 16 + l16;
        aij[((long)t * BATCH + b) * QS + o] = acc[mt][nf][j];
      }
    }
}

// -----------------------------------------------------------------------------
// Kernel 3: masked softmax over t + attend. One block per batch b, one thread
// per o column. 3 coalesced passes over t; passes 2-3 hit L2 (a_ij = 134MB < 192MB).
// -----------------------------------------------------------------------------
__global__ __launch_bounds__(256) void k3_softmax(const float* __restrict__ aij,
                                                  const float* __restrict__ kmask,
                                                  const float* __restrict__ v,
                                                  float* __restrict__ eij,
                                                  float* __restrict__ attend) {
  const int  b   = blockIdx.x;
  const int  o   = threadIdx.x;
  const long cs  = (long)BATCH * QS;            // t stride
  const float* a  = aij + (long)b * QS + o;
  const float* vp = v   + (long)b * QS + o;

  float mx = -FLT_MAX;
  for (int t = 0; t < KV_LEN; ++t) {
    const float msk = kmask[(long)t * BATCH + b];  // uniform -> scalar load
    const float x   = a[(long)t * cs];
    if (msk > 0.f) mx = fmaxf(mx, x);
  }
  float s = 0.f;
  for (int t = 0; t < KV_LEN; ++t) {
    const float msk = kmask[(long)t * BATCH + b];
    const float x   = a[(long)t * cs];
    if (msk > 0.f) s += __expf(x - mx);
  }
  const float rs = (s > 0.f) ? 1.f / s : 0.f;

  float accv = 0.f;
  for (int t = 0; t < KV_LEN; ++t) {
    const float msk = kmask[(long)t * BATCH + b];
    const float x   = a[(long)t * cs];
    const float e   = (msk > 0.f) ? __expf(x - mx) * rs : 0.f;
    eij[(long)t * cs + (long)b * QS + o] = e;
    accv += e * vp[(long)t * cs];
  }
  attend[(long)b * QS + o] = accv;
}

// -----------------------------------------------------------------------------
extern "C" void kernel_launch(void* const* d_in, const int* in_sizes, int n_in,
                              void* d_out, int out_size, void* d_ws,
                              size_t ws_size, hipStream_t stream) {
  const float* q     = (const float*)d_in[0];
  const float* k     = (const float*)d_in[1];
  const float* v     = (const float*)d_in[2];
  const float* kmask = (const float*)d_in[3];
  const float* W     = (const float*)d_in[4];
  const float* bias  = (const float*)d_in[5];

  float* aij    = (float*)d_out;
  float* eij    = aij + (long)KV_LEN * BATCH * QS;
  float* attend = eij + (long)KV_LEN * BATCH * QS;

  // Scratch: Kern[b][o][r], 64*256*768 f32 = 48 MiB
  float* kern = (float*)d_ws;

  k1_qw   <<<dim3(3, 256), 256, 0, stream>>>(q, W, bias, kern);
  k2_attn <<<dim3(16, 64), 256, 0, stream>>>(k, kern, aij);
  k3_softmax<<<64, 256, 0, stream>>>(aij, kmask, v, eij, attend);
}